// WayPointMapping_7378753814753
// MI455X (gfx1250) — compile-verified
//
#include <hip/hip_runtime.h>
#include <hip/hip_bf16.h>

// ---------------- problem constants (from reference) ----------------
constexpr int B   = 2;
constexpr int NC  = 6;
constexpr int CF  = 80;
constexpr int FH  = 32;
constexpr int FW  = 88;
constexpr int NP  = 150000;   // points per batch (divisible by 16)
constexpr int BEVD = 180;
constexpr float IMG_H = 256.0f;
constexpr float IMG_W = 704.0f;

typedef __attribute__((ext_vector_type(2))) float v2f;
typedef __attribute__((ext_vector_type(8))) float v8f;

// ---------------- small matrix helpers (device) ----------------
__device__ __forceinline__ void inv3(const float* m, float* r) {
  float a = m[0], b = m[1], c = m[2];
  float d = m[3], e = m[4], f = m[5];
  float g = m[6], h = m[7], i = m[8];
  float A =  (e * i - f * h);
  float Bc = -(d * i - f * g);
  float C =  (d * h - e * g);
  float det = a * A + b * Bc + c * C;
  float s = 1.0f / det;
  r[0] = A * s;               r[1] = -(b * i - c * h) * s; r[2] =  (b * f - c * e) * s;
  r[3] = Bc * s;              r[4] =  (a * i - c * g) * s; r[5] = -(a * f - c * d) * s;
  r[6] = C * s;               r[7] = -(a * h - b * g) * s; r[8] =  (a * e - b * d) * s;
}

__device__ __forceinline__ void mm3(const float* x, const float* y, float* o) {
  for (int i = 0; i < 3; ++i)
    for (int j = 0; j < 3; ++j)
      o[i * 3 + j] = x[i * 3 + 0] * y[0 * 3 + j] + x[i * 3 + 1] * y[1 * 3 + j] + x[i * 3 + 2] * y[2 * 3 + j];
}

// ---------------- 0) per-(b,c) transform parameter block (36 floats) --------
// layout: [0..8]=A1 (world->cam 3x3), [9..11]=b1, [12..20]=A2 (inv scaled),
//         [21..23]=b2, [24..32]=G (img_aug 3x3), [33..35]=g (img_aug t)
__global__ void bev_setup_params(const float* __restrict__ l2i,
                                 const float* __restrict__ iam,
                                 const float* __restrict__ lam,
                                 float* __restrict__ prm) {
  int t = threadIdx.x;
  if (t >= B * NC) return;
  int b = t / NC, c = t % NC;
  const float* L = l2i + (size_t)t * 16;
  const float* G = iam + (size_t)t * 16;
  const float* A = lam + (size_t)b * 16;

  float lam3[9], lamT[3], invA[9];
  for (int i = 0; i < 3; ++i) {
    for (int j = 0; j < 3; ++j) lam3[i * 3 + j] = A[i * 4 + j];
    lamT[i] = A[i * 4 + 3];
  }
  inv3(lam3, invA);

  float L3[9], Lt[3];
  for (int i = 0; i < 3; ++i) {
    for (int j = 0; j < 3; ++j) L3[i * 3 + j] = L[i * 4 + j];
    Lt[i] = L[i * 4 + 3];
  }

  // cam = L3 * invA * (p - lamT) + Lt  ==  A1 * p + b1
  float A1[9];
  mm3(L3, invA, A1);
  float b1[3];
  for (int i = 0; i < 3; ++i)
    b1[i] = Lt[i] - (A1[i * 3 + 0] * lamT[0] + A1[i * 3 + 1] * lamT[1] + A1[i * 3 + 2] * lamT[2]);

  // faithful scale_intrinsics bug: cam0 rows 0,2 /8; cam1 rows 1,2 /64
  float s[3] = {1.f, 1.f, 1.f};
  if (c == 0) { s[0] = 0.125f; s[2] = 0.125f; }
  else if (c == 1) { s[1] = 1.0f / 64.0f; s[2] = 1.0f / 64.0f; }
  float M[9], ts[3];
  for (int i = 0; i < 3; ++i) {
    for (int j = 0; j < 3; ++j) M[i * 3 + j] = L3[i * 3 + j] * s[i];
    ts[i] = Lt[i] * s[i];
  }
  float A2[9];
  inv3(M, A2);
  float b2[3];
  for (int i = 0; i < 3; ++i)
    b2[i] = -(A2[i * 3 + 0] * ts[0] + A2[i * 3 + 1] * ts[1] + A2[i * 3 + 2] * ts[2]);

  float* P = prm + (size_t)t * 36;
  for (int i = 0; i < 9; ++i) P[i] = A1[i];
  for (int i = 0; i < 3; ++i) P[9 + i] = b1[i];
  for (int i = 0; i < 9; ++i) P[12 + i] = A2[i];
  for (int i = 0; i < 3; ++i) P[21 + i] = b2[i];
  for (int i = 0; i < 3; ++i)
    for (int j = 0; j < 3; ++j) P[24 + i * 3 + j] = G[i * 4 + j];
  for (int i = 0; i < 3; ++i) P[33 + i] = G[i * 4 + 3];
}

// ---------------- 1) zero the channel-last BEV accumulator -----------------
__global__ void bev_zero(float* __restrict__ p, size_t n) {
  size_t i = (size_t)blockIdx.x * blockDim.x + threadIdx.x;
  if (i < n) p[i] = 0.0f;
}

// ---------------- 2) img_feat (B,NC,CF,FH,FW) -> (B,NC,FH,FW,CF) -----------
// LDS 16x16 tile transpose: both global sides coalesced.
__global__ void bev_feat_transpose(const float* __restrict__ in,
                                   float* __restrict__ ft) {
  __shared__ float tile[16][17];
  const int bch = blockIdx.y;            // bc*FH + h
  const int bc = bch / FH, h = bch % FH;
  const int tw = (blockIdx.x % 6) * 16;  // w tile base  (ceil(88/16)=6)
  const int tc = (blockIdx.x / 6) * 16;  // ch tile base (80/16=5)
  const int tx = threadIdx.x, ty = threadIdx.y;

  int ch = tc + ty, w = tw + tx;
  if (ch < CF && w < FW)
    tile[ty][tx] = in[(((size_t)bc * CF + ch) * FH + h) * FW + w];  // coalesced (w)
  __syncthreads();
  int w2 = tw + ty, ch2 = tc + tx;
  if (ch2 < CF && w2 < FW)
    ft[(((size_t)bc * FH + h) * FW + w2) * CF + ch2] = tile[tx][ty]; // coalesced (ch)
}

// ---------------- 3) main: WMMA point transforms + coalesced scatter -------
__global__ void __launch_bounds__(256)
bev_project_scatter(const float4* __restrict__ pts,
                    const float* __restrict__ prm,
                    const float* __restrict__ ft,
                    float* __restrict__ bev) {
  const int lane = threadIdx.x & 31;
  const int wave = threadIdx.x >> 5;
  const int bc = blockIdx.y;
  const int b = bc / NC;
  const int r = lane & 15;          // row (A) / point slot (B)
  const bool hi = lane >= 16;       // upper half-wave supplies K=2,3 of B

  const float* P = prm + (size_t)bc * 36;

  // A operands (16x4 f32, ISA layout): lanes 0-15 -> cols K=0,1 of row r;
  // lanes 16-31 -> cols K=2 and K=3 (translation) of row r. Rows >=3 are zero.
  v2f a1 = {0.0f, 0.0f}, a2 = {0.0f, 0.0f};
  if (r < 3) {
    if (!hi) {
      a1.x = P[r * 3 + 0];      a1.y = P[r * 3 + 1];
      a2.x = P[12 + r * 3 + 0]; a2.y = P[12 + r * 3 + 1];
    } else {
      a1.x = P[r * 3 + 2];      a1.y = P[9 + r];
      a2.x = P[12 + r * 3 + 2]; a2.y = P[21 + r];
    }
  }
  const float g00 = P[24], g01 = P[25], g02 = P[26];
  const float g10 = P[27], g11 = P[28], g12 = P[29];
  const float gt0 = P[33], gt1 = P[34];

  const float* feat = ft + (size_t)bc * (FH * FW * CF);
  float* bv = bev + (size_t)b * ((size_t)BEVD * BEVD * CF);
  const size_t pbase = (size_t)b * NP;

  const int chunks = NP / 16;
  const int wstride = gridDim.x * (blockDim.x >> 5);

  for (int chunk = blockIdx.x * (blockDim.x >> 5) + wave; chunk < chunks; chunk += wstride) {
    // 16 points per wave-iteration; both half-waves read the same 16 points.
    float4 pt = pts[pbase + (size_t)chunk * 16 + r];
    int nxt = chunk + wstride;
    if (nxt < chunks)
      __builtin_prefetch(&pts[pbase + (size_t)nxt * 16 + r], 0, 3);  // near-cache prefetch

    // B operand: lanes 0-15: (x,y); lanes 16-31: (z,1)  -> homogeneous K=4
    v2f bB;
    bB.x = hi ? pt.z : pt.x;
    bB.y = hi ? 1.0f : pt.y;
    v8f cz = {};

    // cam = [A1|b1] * (x,y,z,1)  -- rows 0..2 land in lanes 0..15
    v8f d1 = __builtin_amdgcn_wmma_f32_16x16x4_f32(false, a1, false, bB,
                                                   (short)0, cz, false, false);
    float dep = fminf(fmaxf(d1[2], 1e-5f), 1e5f);
    // dep is clamped well away from 0/inf: single v_rcp_f32 is plenty.
    float inv = __builtin_amdgcn_rcpf(dep);
    float u0 = d1[0] * inv, v0 = d1[1] * inv;
    // img_aug (per-(b,c) uniform, scalar-cached)
    float px = g00 * u0 + g01 * v0 + g02 * dep + gt0;
    float py = g10 * u0 + g11 * v0 + g12 * dep + gt1;
    float c0 = py, c1 = px;  // reference's [1,0] swap
    bool on_img = (c0 >= 0.0f) & (c0 < IMG_H) & (c1 >= 0.0f) & (c1 < IMG_W);
    float uu = c0 * 0.125f, vv = c1 * 0.125f;

    // B operand for unprojection: lanes 0-15 use their OWN (u*d, v*d);
    // lanes 16-31 only need dep from lane-16 -> a single xor-16 shuffle.
    float ud = uu * dep, vd = vv * dep;
    float dp_x = __shfl_xor(dep, 16);
    v2f bW;
    bW.x = hi ? dp_x : ud;
    bW.y = hi ? 1.0f : vd;

    // xyz = [A2|b2] * (u*d, v*d, d, 1)
    v8f d2 = __builtin_amdgcn_wmma_f32_16x16x4_f32(false, a2, false, bW,
                                                   (short)0, cz, false, false);
    float xl = truncf(d2[0]);
    float yl = truncf(d2[1]);
    float zc = d2[2];

    bool valid = on_img & (zc >= 0.0f) &
                 (xl >= -54.0f) & (yl >= -54.0f) & (xl < 54.0f) & (yl < 54.0f) &
                 (uu >= 0.0f) & (uu < (float)FH) & (vv >= 0.0f) & (vv < (float)FW) &
                 (!hi);

    int ui = (int)truncf(uu);
    int vi = (int)truncf(vv);
    int xi = (int)truncf(xl * (1.0f / 0.3f)); xi += (xi < 0) ? BEVD : 0;
    int yi = (int)truncf(yl * (1.0f / 0.3f)); yi += (yi < 0) ? BEVD : 0;

    unsigned mask = (unsigned)__ballot(valid);
    int pack = (ui << 24) | (vi << 16) | (xi << 8) | yi;

    // Whole wave cooperates per valid point. `p` is wave-uniform, so use
    // v_readlane -> scalar broadcast (no DS roundtrip, scalar base address).
    while (mask) {
      int p = __ffs(mask) - 1;
      mask &= mask - 1;
      int pk = __builtin_amdgcn_readlane(pack, p);
      int pui = (pk >> 24) & 0xff, pvi = (pk >> 16) & 0xff;
      int pxi = (pk >> 8) & 0xff,  pyi = pk & 0xff;
      const float* src = feat + (size_t)(pui * FW + pvi) * CF;
      float* dst = bv + (size_t)(pxi * BEVD + pyi) * CF;
#pragma unroll
      for (int t = 0; t < 3; ++t) {
        int ch = lane + t * 32;
        if (ch < CF) atomicAdd(dst + ch, src[ch]);  // coalesced global_atomic_add_f32
      }
    }
  }
}

// ---------------- 4) channel-last BEV -> (B,C,180,180) output --------------
// LDS 16x16 tile transpose of the (xy, ch) matrix per batch.
__global__ void bev_out_transpose(const float* __restrict__ bev,
                                  float* __restrict__ out) {
  __shared__ float tile[16][17];
  const int b = blockIdx.z;
  const int txy = blockIdx.x * 16;   // xy tile base (ceil(32400/16)=2025)
  const int tch = blockIdx.y * 16;   // ch tile base (80/16=5)
  const int tx = threadIdx.x, ty = threadIdx.y;

  int xy = txy + ty, ch = tch + tx;
  if (xy < BEVD * BEVD && ch < CF)
    tile[ty][tx] = bev[((size_t)b * BEVD * BEVD + xy) * CF + ch];      // coalesced (ch)
  __syncthreads();
  int ch2 = tch + ty, xy2 = txy + tx;
  if (xy2 < BEVD * BEVD && ch2 < CF)
    out[((size_t)b * CF + ch2) * (BEVD * BEVD) + xy2] = tile[tx][ty];  // coalesced (xy)
}

// ---------------------------------------------------------------------------
extern "C" void kernel_launch(void* const* d_in, const int* in_sizes, int n_in,
                              void* d_out, int out_size, void* d_ws, size_t ws_size,
                              hipStream_t stream) {
  const float* img_feat = (const float*)d_in[0];  // (B,NC,CF,FH,FW)
  const float* points   = (const float*)d_in[1];  // (B,N,4)
  const float* l2i      = (const float*)d_in[2];  // (B,NC,4,4)
  const float* iam      = (const float*)d_in[3];  // (B,NC,4,4)
  const float* lam      = (const float*)d_in[4];  // (B,4,4)
  float* out = (float*)d_out;

  float* ws  = (float*)d_ws;
  float* prm = ws;                                       // 432 floats (pad 512)
  float* ft  = ws + 512;                                 // B*NC*FH*FW*CF
  float* bev = ft + (size_t)B * NC * FH * FW * CF;       // B*BEV*BEV*CF

  bev_setup_params<<<1, 32, 0, stream>>>(l2i, iam, lam, prm);

  {
    size_t n = (size_t)B * BEVD * BEVD * CF;
    bev_zero<<<(int)((n + 255) / 256), 256, 0, stream>>>(bev, n);
  }
  {
    dim3 grid(6 * 5, B * NC * FH);   // (w tiles x ch tiles, bc*h)
    dim3 block(16, 16);
    bev_feat_transpose<<<grid, block, 0, stream>>>(img_feat, ft);
  }
  {
    dim3 grid(128, B * NC);
    bev_project_scatter<<<grid, 256, 0, stream>>>((const float4*)points, prm, ft, bev);
  }
  {
    dim3 grid((BEVD * BEVD + 15) / 16, CF / 16, B);
    dim3 block(16, 16);
    bev_out_transpose<<<grid, block, 0, stream>>>(bev, out);
  }
}